// Model_3315714752591
// MI455X (gfx1250) — compile-verified
//
#include <hip/hip_runtime.h>

typedef __attribute__((ext_vector_type(2))) float v2f;
typedef __attribute__((ext_vector_type(8))) float v8f;
typedef int v4i __attribute__((vector_size(16)));

#define N_NODES 50000
#define N_EDGES 800000
#define N_LABEL 200000
#define HID     128
#define EDGE_DIM 32

#define AS_STRIDE 132   // 16x128 A tile, padded: 132 % 64 == 4 -> conflict-free
#define ZS_STRIDE 260   // 16x256 A tile, padded: 260 % 64 == 4 -> conflict-free

#if __has_builtin(__builtin_amdgcn_global_load_async_to_lds_b128) && \
    __has_builtin(__builtin_amdgcn_s_wait_asynccnt)
#define GINE_ASYNC_LDS 1
#endif

// Async global->LDS 16B copy (falls back to a plain float4 copy).
__device__ __forceinline__ void g2l_b128(const float* g, float* l) {
#ifdef GINE_ASYNC_LDS
    __builtin_amdgcn_global_load_async_to_lds_b128(
        (__attribute__((address_space(1))) v4i*)g,
        (__attribute__((address_space(3))) v4i*)l, 0, 0);
#else
    *reinterpret_cast<float4*>(l) = *reinterpret_cast<const float4*>(g);
#endif
}

__device__ __forceinline__ void g2l_wait() {
#ifdef GINE_ASYNC_LDS
    __builtin_amdgcn_s_wait_asynccnt(0);
#endif
}

// ---------------------------------------------------------------------------
// Zero a float buffer (segment-sum accumulator init)
// ---------------------------------------------------------------------------
__global__ void gine_zero_kernel(float* __restrict__ p, int n) {
    int i = blockIdx.x * blockDim.x + threadIdx.x;
    int stride = gridDim.x * blockDim.x;
    for (; i < n; i += stride) p[i] = 0.0f;
}

// ---------------------------------------------------------------------------
// Fused message + aggregate:
//   m = relu(x[src] + edge_attr @ We + be);  agg[dst] += m   (atomic f32)
// One wave per edge (e is wave-uniform -> scalar loads for edge_attr/indices).
// ---------------------------------------------------------------------------
__global__ __launch_bounds__(256) void gine_msg_kernel(
    const float* __restrict__ x, const int* __restrict__ edge_index,
    const float* __restrict__ edge_attr, const float* __restrict__ We,
    const float* __restrict__ be, float* __restrict__ agg) {
    __shared__ float Ws[EDGE_DIM * HID];
    __shared__ float Bs[HID];
    int tid = threadIdx.x;
    for (int i = tid * 4; i < EDGE_DIM * HID; i += 256 * 4) g2l_b128(&We[i], &Ws[i]);
    if (tid < HID) Bs[tid] = be[tid];
    g2l_wait();
    __syncthreads();

    int lane = tid & 31;
    int wave = tid >> 5;
    int e = blockIdx.x * 8 + wave;           // N_EDGES % 8 == 0
    int src = edge_index[e];
    int dst = edge_index[N_EDGES + e];
    int c0 = lane * 4;                       // each lane owns 4 contiguous channels

    const float4 xv = *reinterpret_cast<const float4*>(x + (long)src * HID + c0);
    float m0 = Bs[c0 + 0], m1 = Bs[c0 + 1], m2 = Bs[c0 + 2], m3 = Bs[c0 + 3];

    const float* ea = edge_attr + (long)e * EDGE_DIM;   // wave-uniform address
#pragma unroll 8
    for (int k = 0; k < EDGE_DIM; ++k) {
        float a = ea[k];
        const float* wrow = &Ws[k * HID + c0];
        m0 = fmaf(a, wrow[0], m0);
        m1 = fmaf(a, wrow[1], m1);
        m2 = fmaf(a, wrow[2], m2);
        m3 = fmaf(a, wrow[3], m3);
    }
    m0 = fmaxf(m0 + xv.x, 0.0f);
    m1 = fmaxf(m1 + xv.y, 0.0f);
    m2 = fmaxf(m2 + xv.z, 0.0f);
    m3 = fmaxf(m3 + xv.w, 0.0f);

    float* ap = agg + (long)dst * HID + c0;
    atomicAdd(ap + 0, m0);
    atomicAdd(ap + 1, m1);
    atomicAdd(ap + 2, m2);
    atomicAdd(ap + 3, m3);
}

// ---------------------------------------------------------------------------
// Node linear: out = [(x + agg) @ W + b], optional relu.
// Block = 256 threads (8 waves) -> 16 rows; wave w owns a 16-column tile.
// A (x+agg, 16x128) and the full 128x128 W staged in LDS (W via async copy).
// Inner loop is pure DS + V_WMMA_F32_16X16X4_F32 (32 K-steps).
// ---------------------------------------------------------------------------
__global__ __launch_bounds__(256) void gine_node_wmma_kernel(
    const float* __restrict__ x, const float* __restrict__ agg,
    const float* __restrict__ W, const float* __restrict__ bias,
    float* __restrict__ out, int do_relu) {
    __shared__ float Ws[HID * HID];          // 64 KB, async-staged
    __shared__ float As[16 * AS_STRIDE];     // padded A tile
    int tid = threadIdx.x;

    for (int i = tid * 4; i < HID * HID; i += 256 * 4) g2l_b128(&W[i], &Ws[i]);

    long base = (long)blockIdx.x * (16 * HID);
    for (int i = tid; i < 16 * HID; i += 256) {
        int row = i >> 7, c = i & (HID - 1);
        As[row * AS_STRIDE + c] = x[base + i] + agg[base + i];
    }
    g2l_wait();
    __syncthreads();

    int lane = tid & 31;
    int wave = tid >> 5;
    int lo = lane & 15;
    int hi = lane >> 4;
    int col = wave * 16 + lo;

    v8f c = {};
    for (int k = 0; k < HID; k += 4) {
        int ka = k + hi * 2;                 // lanes 16-31 hold K+2, K+3
        v2f a, b;
        a.x = As[lo * AS_STRIDE + ka];
        a.y = As[lo * AS_STRIDE + ka + 1];
        b.x = Ws[ka * HID + col];
        b.y = Ws[(ka + 1) * HID + col];
        c = __builtin_amdgcn_wmma_f32_16x16x4_f32(false, a, false, b,
                                                  (short)0, c, false, false);
    }

    float bv = bias[col];
#pragma unroll
    for (int v = 0; v < 8; ++v) {
        float val = c[v] + bv;
        if (do_relu) val = fmaxf(val, 0.0f);
        out[base + (long)(hi * 8 + v) * HID + col] = val;  // C: VGPR v -> row v / v+8
    }
}

// ---------------------------------------------------------------------------
// Edge classifier: z = relu(concat(x[a], x[b]) @ W1 + b1); pred = z @ W2 + b2
// Block = 256 threads (8 waves), each wave owns 16 labels.
// Full W1 (256x128 = 128 KB) async-staged once per block; per-wave gathered
// 16x256 A tiles in padded LDS (8 x 16.25 KB). ~259 KB LDS of the 320 KB WGP.
// Second linear (128->1) evaluated in-register on the WMMA C layout.
// ---------------------------------------------------------------------------
__global__ __launch_bounds__(256) void gine_cls_wmma_kernel(
    const float* __restrict__ xf, const int* __restrict__ eli,
    const float* __restrict__ W1, const float* __restrict__ b1,
    const float* __restrict__ W2, const float* __restrict__ b2,
    float* __restrict__ pred) {
    __shared__ float W1s[256 * HID];             // 128 KB, async-staged
    __shared__ float Zs[8 * 16 * ZS_STRIDE];     // 8 padded per-wave tiles
    int tid = threadIdx.x;
    int lane = tid & 31;
    int wave = tid >> 5;

    for (int i = tid * 4; i < 256 * HID; i += 256 * 4) g2l_b128(&W1[i], &W1s[i]);

    int tile = blockIdx.x * 8 + wave;
    bool active = tile < (N_LABEL / 16);
    int t0 = tile * 16;
    float* Zw = Zs + wave * 16 * ZS_STRIDE;

    if (active) {
        for (int i = lane; i < 16 * 256; i += 32) {
            int row = i >> 8;
            int c = i & 255;
            int node = (c < HID) ? eli[t0 + row] : eli[N_LABEL + t0 + row];
            Zw[row * ZS_STRIDE + c] = xf[(long)node * HID + (c & (HID - 1))];
        }
    }
    g2l_wait();
    __syncthreads();
    if (!active) return;

    int lo = lane & 15;
    int hi = lane >> 4;

    v8f acc[8] = {};
    for (int k = 0; k < 256; k += 4) {
        int ka = k + hi * 2;
        v2f a;
        a.x = Zw[lo * ZS_STRIDE + ka];
        a.y = Zw[lo * ZS_STRIDE + ka + 1];
#pragma unroll
        for (int nt = 0; nt < 8; ++nt) {
            int col = nt * 16 + lo;
            v2f b;
            b.x = W1s[ka * HID + col];
            b.y = W1s[(ka + 1) * HID + col];
            acc[nt] = __builtin_amdgcn_wmma_f32_16x16x4_f32(false, a, false, b,
                                                            (short)0, acc[nt],
                                                            false, false);
        }
    }

    // bias + relu + (z @ W2): lane l holds column nt*16 + l%16 of rows v + (l/16)*8
    float part[8];
#pragma unroll
    for (int v = 0; v < 8; ++v) part[v] = 0.0f;
#pragma unroll
    for (int nt = 0; nt < 8; ++nt) {
        int col = nt * 16 + lo;
        float bb = b1[col];
        float w2 = W2[col];
#pragma unroll
        for (int v = 0; v < 8; ++v) {
            float z = fmaxf(acc[nt][v] + bb, 0.0f);
            part[v] = fmaf(z, w2, part[v]);
        }
    }
    // reduce across the 16 lanes that share a row (xor masks 1,2,4,8 stay in-half)
#pragma unroll
    for (int v = 0; v < 8; ++v) {
        for (int off = 8; off >= 1; off >>= 1)
            part[v] += __shfl_xor(part[v], off, 32);
    }
    if (lo == 0) {
        float bias2 = b2[0];
#pragma unroll
        for (int v = 0; v < 8; ++v)
            pred[t0 + hi * 8 + v] = part[v] + bias2;
    }
}

// ---------------------------------------------------------------------------
extern "C" void kernel_launch(void* const* d_in, const int* in_sizes, int n_in,
                              void* d_out, int out_size, void* d_ws, size_t ws_size,
                              hipStream_t stream) {
    const float* x    = (const float*)d_in[0];
    const int*   ei   = (const int*)d_in[1];
    const float* eatt = (const float*)d_in[2];
    const int*   eli  = (const int*)d_in[3];
    const float* leW  = (const float*)d_in[4];
    const float* leB  = (const float*)d_in[5];
    const float* nnW  = (const float*)d_in[6];
    const float* nnB  = (const float*)d_in[7];
    const float* cW1  = (const float*)d_in[8];
    const float* cB1  = (const float*)d_in[9];
    const float* cW2  = (const float*)d_in[10];
    const float* cB2  = (const float*)d_in[11];

    float* out  = (float*)d_out;
    float* pred = out;                       // [N_LABEL]
    float* x2   = out + N_LABEL;             // [N_NODES, HID] final node features

    float* ws  = (float*)d_ws;
    float* agg = ws;                         // [N_NODES, HID]
    float* x1  = ws + (size_t)N_NODES * HID; // [N_NODES, HID]

    const int nAgg = N_NODES * HID;

    // ---- layer 0 ----
    gine_zero_kernel<<<(nAgg + 255) / 256, 256, 0, stream>>>(agg, nAgg);
    gine_msg_kernel<<<N_EDGES / 8, 256, 0, stream>>>(x, ei, eatt, leW, leB, agg);
    gine_node_wmma_kernel<<<N_NODES / 16, 256, 0, stream>>>(x, agg, nnW, nnB, x1, 1);

    // ---- layer 1 ----
    gine_zero_kernel<<<(nAgg + 255) / 256, 256, 0, stream>>>(agg, nAgg);
    gine_msg_kernel<<<N_EDGES / 8, 256, 0, stream>>>(x1, ei, eatt, leW, leB, agg);
    gine_node_wmma_kernel<<<N_NODES / 16, 256, 0, stream>>>(x1, agg, nnW, nnB, x2, 0);

    // ---- edge classifier ----
    int clsBlocks = (N_LABEL / 16 + 7) / 8;
    gine_cls_wmma_kernel<<<clsBlocks, 256, 0, stream>>>(x2, eli, cW1, cB1, cW2,
                                                        cB2, pred);
}